// semiCRF_34153579938261
// MI455X (gfx1250) — compile-verified
//
#include <hip/hip_runtime.h>

#define L_LEN 2048
#define T_TAG 8
#define B_BATCH 256
#define NEG_INIT -10000.0f

typedef float v8f __attribute__((ext_vector_type(8)));
typedef float v2f __attribute__((ext_vector_type(2)));

#if defined(__HIP_DEVICE_COMPILE__) && __has_builtin(__builtin_amdgcn_wmma_f32_16x16x4_f32)
#define HAS_WMMA_F32X4 1
#else
#define HAS_WMMA_F32X4 0
#endif

struct MatCtx {
#if HAS_WMMA_F32X4
  v2f A1, A2;   // exp(trans - c) in WMMA A layout (rows 8..15 zero)
#else
  float P[64];  // fallback: per-lane replicated exp(trans - c)
#endif
};

__device__ __forceinline__ float bperm_lo(float v, int idxbyte) {
#if defined(__HIP_DEVICE_COMPILE__)
  return __int_as_float(__builtin_amdgcn_ds_bpermute(idxbyte, __float_as_int(v)));
#else
  (void)idxbyte; return v;
#endif
}

__device__ __forceinline__ float fast_rcp(float x) {
#if defined(__HIP_DEVICE_COMPILE__) && __has_builtin(__builtin_amdgcn_rcpf)
  return __builtin_amdgcn_rcpf(x);
#else
  return 1.0f / x;
#endif
}

// out[i] = sum_j M[i][j] * in[j] for 8x8 positive matrix (encoded in ctx),
// in[] lives per-lane (lane = batch, only lanes 0..15 meaningful).
__device__ __forceinline__ void matvec8(const MatCtx& ctx, const float* in, float* out,
                                        int permbyte, bool lo) {
#if HAS_WMMA_F32X4
  // Build B (K x N = tag x batch): VGPR0 = {K0 | K2}, VGPR1 = {K1 | K3} per half-wave.
  float p2 = bperm_lo(in[2], permbyte);
  float p3 = bperm_lo(in[3], permbyte);
  float p6 = bperm_lo(in[6], permbyte);
  float p7 = bperm_lo(in[7], permbyte);
  v2f B1, B2;
  B1.x = lo ? in[0] : p2;
  B1.y = lo ? in[1] : p3;
  B2.x = lo ? in[4] : p6;
  B2.y = lo ? in[5] : p7;
  v8f d = {0.f, 0.f, 0.f, 0.f, 0.f, 0.f, 0.f, 0.f};
  d = __builtin_amdgcn_wmma_f32_16x16x4_f32(false, ctx.A1, false, B1, (short)0, d, false, false);
  d = __builtin_amdgcn_wmma_f32_16x16x4_f32(false, ctx.A2, false, B2, (short)0, d, false, false);
#pragma unroll
  for (int k = 0; k < 8; ++k) out[k] = d[k];
#else
#pragma unroll
  for (int i = 0; i < 8; ++i) {
    float s = 0.f;
#pragma unroll
    for (int j = 0; j < 8; ++j) s = fmaf(ctx.P[i * 8 + j], in[j], s);
    out[i] = s;
  }
  (void)permbyte; (void)lo;
#endif
}

// ---------------- forward ----------------
// Slot state per span-age m: mv[8] = matvec(exp(alpha-mx)), mx = scale.
// alpha_pos = log( sum_m mv_m[t'] * exp(e_m[t'] + mx_m - g) ) + g + c
#define FSTEP(POS, HD, H1, H2, H3, MV0, MV1, MV2, MV3, MX0, MX1, MX2, MX3)          \
  do {                                                                              \
    const float* rp = fb + (size_t)(POS)*T_TAG;                                     \
    __builtin_prefetch(rp + 8 * T_TAG, 0, 0);                                       \
    const float mk = mb[POS];                                                       \
    float4 r0 = *(const float4*)(rp);                                               \
    float4 r1 = *(const float4*)(rp + 4);                                           \
    HD[0] = r0.x * mk; HD[1] = r0.y * mk; HD[2] = r0.z * mk; HD[3] = r0.w * mk;     \
    HD[4] = r1.x * mk; HD[5] = r1.y * mk; HD[6] = r1.z * mk; HD[7] = r1.w * mk;     \
    float g = fmaxf(fmaxf(MX0, MX1), fmaxf(MX2, MX3));                              \
    float d0 = MX0 - g, d1 = MX1 - g, d2 = MX2 - g, d3 = MX3 - g;                   \
    float acc[8];                                                                   \
    _Pragma("unroll") for (int k = 0; k < 8; ++k) {                                 \
      float e0 = HD[k];                                                             \
      float e1 = fmaxf(e0, H1[k]);                                                  \
      float e2 = fmaxf(e1, H2[k]);                                                  \
      float e3 = fmaxf(e2, H3[k]);                                                  \
      float s = MV0[k] * __expf(e0 + d0);                                           \
      s = fmaf(MV1[k], __expf(e1 + d1), s);                                         \
      s = fmaf(MV2[k], __expf(e2 + d2), s);                                         \
      s = fmaf(MV3[k], __expf(e3 + d3), s);                                         \
      acc[k] = s;                                                                   \
    }                                                                               \
    float amax = acc[0];                                                            \
    _Pragma("unroll") for (int k = 1; k < 8; ++k) amax = fmaxf(amax, acc[k]);       \
    float rin = fast_rcp(amax);                                                     \
    float ebn[8];                                                                   \
    _Pragma("unroll") for (int k = 0; k < 8; ++k) ebn[k] = acc[k] * rin;            \
    float mxn = g + c + __logf(amax);                                               \
    if (lo) {                                                                       \
      float4 s0, s1;                                                                \
      s0.x = ebn[0]; s0.y = ebn[1]; s0.z = ebn[2]; s0.w = ebn[3];                   \
      s1.x = ebn[4]; s1.y = ebn[5]; s1.z = ebn[6]; s1.w = ebn[7];                   \
      *(float4*)(fout + (size_t)(POS)*T_TAG) = s0;                                  \
      *(float4*)(fout + (size_t)(POS)*T_TAG + 4) = s1;                              \
      gout[POS] = mxn;                                                              \
    }                                                                               \
    matvec8(ctx, ebn, MV3, permbyte, lo);                                           \
    MX3 = mxn;                                                                      \
  } while (0)

__device__ __forceinline__ void run_forward(
    int b0, const float* __restrict__ feats, const float* __restrict__ masks,
    const float* __restrict__ trans,
    float* __restrict__ fexp, float* __restrict__ gf, float* __restrict__ aa) {
  const int lane = (int)threadIdx.x;
  const int tp = lane & 15;
  const bool lo = lane < 16;
  const int permbyte = tp << 2;
  const int b = b0 + tp;
  const float* fb = feats + (size_t)b * (L_LEN * T_TAG);
  const float* mb = masks + (size_t)b * L_LEN;
  float* fout = fexp + (size_t)b * (L_LEN * T_TAG);
  float* gout = gf + (size_t)b * L_LEN;

  float c = trans[0];
  for (int i = 1; i < 64; ++i) c = fmaxf(c, trans[i]);

  MatCtx ctx;
#if HAS_WMMA_F32X4
  {
    const int j0 = lo ? 0 : 2, j1 = lo ? 1 : 3, j2 = lo ? 4 : 6, j3 = lo ? 5 : 7;
    const bool row = tp < 8;  // M(i=t', j=t) = exp(trans[t'][t] - c)
    ctx.A1.x = row ? __expf(trans[tp * 8 + j0] - c) : 0.0f;
    ctx.A1.y = row ? __expf(trans[tp * 8 + j1] - c) : 0.0f;
    ctx.A2.x = row ? __expf(trans[tp * 8 + j2] - c) : 0.0f;
    ctx.A2.y = row ? __expf(trans[tp * 8 + j3] - c) : 0.0f;
  }
#else
#pragma unroll
  for (int i = 0; i < 8; ++i)
#pragma unroll
    for (int j = 0; j < 8; ++j) ctx.P[i * 8 + j] = __expf(trans[i * 8 + j] - c);
#endif

  float hA[8], hB[8], hC[8], hD[8];
  float mvA[8], mvB[8], mvC[8], mvD[8];
  float mxA, mxB, mxC, mxD;

  {  // pos 0: alpha0 = masked feats row 0
    const float mk = mb[0];
    float4 r0 = *(const float4*)(fb);
    float4 r1 = *(const float4*)(fb + 4);
    hA[0] = r0.x * mk; hA[1] = r0.y * mk; hA[2] = r0.z * mk; hA[3] = r0.w * mk;
    hA[4] = r1.x * mk; hA[5] = r1.y * mk; hA[6] = r1.z * mk; hA[7] = r1.w * mk;
#pragma unroll
    for (int k = 0; k < 8; ++k) { hB[k] = hA[k]; hC[k] = hA[k]; hD[k] = hA[k]; }
    float mx0 = hA[0];
#pragma unroll
    for (int k = 1; k < 8; ++k) mx0 = fmaxf(mx0, hA[k]);
    float eb0[8];
#pragma unroll
    for (int k = 0; k < 8; ++k) eb0[k] = __expf(hA[k] - mx0);
    if (lo) {
      float4 s0, s1;
      s0.x = eb0[0]; s0.y = eb0[1]; s0.z = eb0[2]; s0.w = eb0[3];
      s1.x = eb0[4]; s1.y = eb0[5]; s1.z = eb0[6]; s1.w = eb0[7];
      *(float4*)(fout) = s0;
      *(float4*)(fout + 4) = s1;
      gout[0] = mx0;
    }
    matvec8(ctx, eb0, mvA, permbyte, lo);
    mxA = mx0;
#pragma unroll
    for (int k = 0; k < 8; ++k) { mvB[k] = 0.f; mvC[k] = 0.f; mvD[k] = 0.f; }
    mxB = NEG_INIT; mxC = NEG_INIT; mxD = NEG_INIT;
  }

  int pos = 1;
  for (int it = 0; it < 511; ++it) {
    FSTEP(pos,     hD, hA, hB, hC, mvA, mvB, mvC, mvD, mxA, mxB, mxC, mxD);
    FSTEP(pos + 1, hC, hD, hA, hB, mvD, mvA, mvB, mvC, mxD, mxA, mxB, mxC);
    FSTEP(pos + 2, hB, hC, hD, hA, mvC, mvD, mvA, mvB, mxC, mxD, mxA, mxB);
    FSTEP(pos + 3, hA, hB, hC, hD, mvB, mvC, mvD, mvA, mxB, mxC, mxD, mxA);
    pos += 4;
  }
  FSTEP(pos,     hD, hA, hB, hC, mvA, mvB, mvC, mvD, mxA, mxB, mxC, mxD);
  FSTEP(pos + 1, hC, hD, hA, hB, mvD, mvA, mvB, mvC, mxD, mxA, mxB, mxC);
  FSTEP(pos + 2, hB, hC, hD, hA, mvC, mvD, mvA, mvB, mxC, mxD, mxA, mxB);

  if (lo) {  // a[b] = LSE_t f[b, L-1, :]
    const float* fl = fout + (size_t)(L_LEN - 1) * T_TAG;
    float s = fl[0];
#pragma unroll
    for (int k = 1; k < 8; ++k) s += fl[k];
    aa[b] = gout[L_LEN - 1] + __logf(s);
  }
}

// ---------------- backward ----------------
// Slot state per age m: eb[8] = exp(beta-mx), mx.  Emission folds in BEFORE the matvec:
// beta_pos[t] = log( sum_{t'} P[t'][t] * sum_m eb_m[t']*exp(e_m[t'] + mx_m - g) ) + g + c
#define BSTEP(POS, HD, H1, H2, H3, EB0, EB1, EB2, EB3, MX0, MX1, MX2, MX3)          \
  do {                                                                              \
    const int rowp = L_LEN - (POS);                                                 \
    const float* rp = fb + (size_t)rowp * T_TAG;                                    \
    __builtin_prefetch(rp - 8 * T_TAG, 0, 0);                                       \
    const float mk = mb[rowp];                                                      \
    float4 r0 = *(const float4*)(rp);                                               \
    float4 r1 = *(const float4*)(rp + 4);                                           \
    HD[0] = r0.x * mk; HD[1] = r0.y * mk; HD[2] = r0.z * mk; HD[3] = r0.w * mk;     \
    HD[4] = r1.x * mk; HD[5] = r1.y * mk; HD[6] = r1.z * mk; HD[7] = r1.w * mk;     \
    float g = fmaxf(fmaxf(MX0, MX1), fmaxf(MX2, MX3));                              \
    float d0 = MX0 - g, d1 = MX1 - g, d2 = MX2 - g, d3 = MX3 - g;                   \
    float comb[8];                                                                  \
    _Pragma("unroll") for (int k = 0; k < 8; ++k) {                                 \
      float e0 = HD[k];                                                             \
      float e1 = fmaxf(e0, H1[k]);                                                  \
      float e2 = fmaxf(e1, H2[k]);                                                  \
      float e3 = fmaxf(e2, H3[k]);                                                  \
      float s = EB0[k] * __expf(e0 + d0);                                           \
      s = fmaf(EB1[k], __expf(e1 + d1), s);                                         \
      s = fmaf(EB2[k], __expf(e2 + d2), s);                                         \
      s = fmaf(EB3[k], __expf(e3 + d3), s);                                         \
      comb[k] = s;                                                                  \
    }                                                                               \
    float outv[8];                                                                  \
    matvec8(ctx, comb, outv, permbyte, lo);                                         \
    float amax = outv[0];                                                           \
    _Pragma("unroll") for (int k = 1; k < 8; ++k) amax = fmaxf(amax, outv[k]);      \
    float rin = fast_rcp(amax);                                                     \
    _Pragma("unroll") for (int k = 0; k < 8; ++k) EB3[k] = outv[k] * rin;           \
    float mxn = g + c + __logf(amax);                                               \
    MX3 = mxn;                                                                      \
    if (lo) {                                                                       \
      const size_t ol = (size_t)(L_LEN - 1 - (POS));                                \
      float4 s0, s1;                                                                \
      s0.x = EB3[0]; s0.y = EB3[1]; s0.z = EB3[2]; s0.w = EB3[3];                   \
      s1.x = EB3[4]; s1.y = EB3[5]; s1.z = EB3[6]; s1.w = EB3[7];                   \
      *(float4*)(bout + ol * T_TAG) = s0;                                           \
      *(float4*)(bout + ol * T_TAG + 4) = s1;                                       \
      gbout[ol] = mxn;                                                              \
    }                                                                               \
  } while (0)

__device__ __forceinline__ void run_backward(
    int b0, const float* __restrict__ feats, const float* __restrict__ masks,
    const float* __restrict__ trans,
    float* __restrict__ bexp, float* __restrict__ gb) {
  const int lane = (int)threadIdx.x;
  const int tp = lane & 15;
  const bool lo = lane < 16;
  const int permbyte = tp << 2;
  const int b = b0 + tp;
  const float* fb = feats + (size_t)b * (L_LEN * T_TAG);
  const float* mb = masks + (size_t)b * L_LEN;
  float* bout = bexp + (size_t)b * (L_LEN * T_TAG);
  float* gbout = gb + (size_t)b * L_LEN;

  float c = trans[0];
  for (int i = 1; i < 64; ++i) c = fmaxf(c, trans[i]);

  MatCtx ctx;
#if HAS_WMMA_F32X4
  {
    const int j0 = lo ? 0 : 2, j1 = lo ? 1 : 3, j2 = lo ? 4 : 6, j3 = lo ? 5 : 7;
    const bool row = tp < 8;  // M(i=t, j=t') = exp(trans[t'][t] - c)  (transposed)
    ctx.A1.x = row ? __expf(trans[j0 * 8 + tp] - c) : 0.0f;
    ctx.A1.y = row ? __expf(trans[j1 * 8 + tp] - c) : 0.0f;
    ctx.A2.x = row ? __expf(trans[j2 * 8 + tp] - c) : 0.0f;
    ctx.A2.y = row ? __expf(trans[j3 * 8 + tp] - c) : 0.0f;
  }
#else
#pragma unroll
  for (int i = 0; i < 8; ++i)
#pragma unroll
    for (int j = 0; j < 8; ++j) ctx.P[i * 8 + j] = __expf(trans[j * 8 + i] - c);
#endif

  float hA[8], hB[8], hC[8], hD[8];
  float ebA[8], ebB[8], ebC[8], ebD[8];
  float mxA, mxB, mxC, mxD;

  {  // beta0 = 0 at scan pos 0  -> output row L-1
    const float mk = mb[L_LEN - 1];
    float4 r0 = *(const float4*)(fb + (size_t)(L_LEN - 1) * T_TAG);
    float4 r1 = *(const float4*)(fb + (size_t)(L_LEN - 1) * T_TAG + 4);
    hA[0] = r0.x * mk; hA[1] = r0.y * mk; hA[2] = r0.z * mk; hA[3] = r0.w * mk;
    hA[4] = r1.x * mk; hA[5] = r1.y * mk; hA[6] = r1.z * mk; hA[7] = r1.w * mk;
#pragma unroll
    for (int k = 0; k < 8; ++k) { hB[k] = hA[k]; hC[k] = hA[k]; hD[k] = hA[k]; }
#pragma unroll
    for (int k = 0; k < 8; ++k) { ebA[k] = 1.f; ebB[k] = 1.f; ebC[k] = 1.f; ebD[k] = 1.f; }
    mxA = 0.f; mxB = NEG_INIT; mxC = NEG_INIT; mxD = NEG_INIT;
    if (lo) {
      float4 one4; one4.x = one4.y = one4.z = one4.w = 1.f;
      *(float4*)(bout + (size_t)(L_LEN - 1) * T_TAG) = one4;
      *(float4*)(bout + (size_t)(L_LEN - 1) * T_TAG + 4) = one4;
      gbout[L_LEN - 1] = 0.f;
    }
  }

  int pos = 1;
  for (int it = 0; it < 511; ++it) {
    BSTEP(pos,     hD, hA, hB, hC, ebA, ebB, ebC, ebD, mxA, mxB, mxC, mxD);
    BSTEP(pos + 1, hC, hD, hA, hB, ebD, ebA, ebB, ebC, mxD, mxA, mxB, mxC);
    BSTEP(pos + 2, hB, hC, hD, hA, ebC, ebD, ebA, ebB, mxC, mxD, mxA, mxB);
    BSTEP(pos + 3, hA, hB, hC, hD, ebB, ebC, ebD, ebA, mxB, mxC, mxD, mxA);
    pos += 4;
  }
  BSTEP(pos,     hD, hA, hB, hC, ebA, ebB, ebC, ebD, mxA, mxB, mxC, mxD);
  BSTEP(pos + 1, hC, hD, hA, hB, ebD, ebA, ebB, ebC, mxD, mxA, mxB, mxC);
  BSTEP(pos + 2, hB, hC, hD, hA, ebC, ebD, ebA, ebB, mxC, mxD, mxA, mxB);
}

__global__ __launch_bounds__(32) void semicrf_scan_kernel(
    const float* __restrict__ feats, const float* __restrict__ masks,
    const float* __restrict__ trans,
    float* __restrict__ fexp, float* __restrict__ gf,
    float* __restrict__ bexp, float* __restrict__ gb, float* __restrict__ aa) {
  const int blk = (int)blockIdx.x;
  if (blk < 16) {
    run_forward(blk * 16, feats, masks, trans, fexp, gf, aa);
  } else {
    run_backward((blk - 16) * 16, feats, masks, trans, bexp, gb);
  }
}

__global__ __launch_bounds__(256) void semicrf_combine_kernel(
    const float* __restrict__ fexp, const float* __restrict__ gf,
    const float* __restrict__ bexp, const float* __restrict__ gb,
    const float* __restrict__ aa, float* __restrict__ out, int total) {
  int idx = (int)(blockIdx.x * blockDim.x + threadIdx.x);
  if (idx >= total) return;
  int bl = idx / T_TAG;          // = b*L + l
  int b = bl / L_LEN;
  // exp(f + b_rev - a) = fexp * bexp * exp(gf + gb - a)
  out[idx] = fexp[idx] * bexp[idx] * __expf(gf[bl] + gb[bl] - aa[b]);
}

extern "C" void kernel_launch(void* const* d_in, const int* in_sizes, int n_in,
                              void* d_out, int out_size, void* d_ws, size_t ws_size,
                              hipStream_t stream) {
  (void)in_sizes; (void)n_in; (void)out_size; (void)ws_size;
  const float* feats = (const float*)d_in[0];
  const float* masks = (const float*)d_in[1];
  const float* trans = (const float*)d_in[2];

  float* fexp = (float*)d_ws;                                   // B*L*T
  float* gf   = fexp + (size_t)B_BATCH * L_LEN * T_TAG;         // B*L
  float* bexp = gf   + (size_t)B_BATCH * L_LEN;                 // B*L*T
  float* gb   = bexp + (size_t)B_BATCH * L_LEN * T_TAG;         // B*L
  float* aa   = gb   + (size_t)B_BATCH * L_LEN;                 // B

  semicrf_scan_kernel<<<dim3(32), dim3(32), 0, stream>>>(
      feats, masks, trans, fexp, gf, bexp, gb, aa);

  const int total = B_BATCH * L_LEN * T_TAG;
  semicrf_combine_kernel<<<dim3((total + 255) / 256), dim3(256), 0, stream>>>(
      fexp, gf, bexp, gb, aa, (float*)d_out, total);
}